// attn_mech_31585189495295
// MI455X (gfx1250) — compile-verified
//
#include <hip/hip_runtime.h>
#include <hip/hip_bf16.h>
#include <math.h>

// ---------------- problem constants (fixed by the reference) ----------------
#define FIN    256   // input features F
#define NHEADS 4     // heads H
#define DHEAD  64    // per-head dim D
#define COLS   256   // H*D

typedef __attribute__((ext_vector_type(2))) float v2f;
typedef __attribute__((ext_vector_type(8))) float v8f;

// ---------------------------------------------------------------------------
// 0) init: zero output accumulator + softmax denom, set segment-max to -inf
// ---------------------------------------------------------------------------
__global__ void gat_init(float* __restrict__ out, float* __restrict__ m,
                         float* __restrict__ denom, long total_out, int total_mh) {
    long idx = (long)blockIdx.x * blockDim.x + threadIdx.x;
    if (idx < total_out) out[idx] = 0.0f;
    if (idx < total_mh) {
        m[idx]     = -__builtin_inff();
        denom[idx] = 0.0f;
    }
}

// ---------------------------------------------------------------------------
// 1) Wh = x @ W  via V_WMMA_F32_16X16X4_F32 (exact fp32 matrix path)
//    grid = (ceil(N/128), NHEADS), block = 128 (4 waves)
//    blockIdx.y selects head h: output columns [h*64, h*64+64)
//    Each wave owns a 32-row x 64-col tile (8 accumulators); K loop of 64
//    steps (K=4 each) -> 8 WMMAs per step, 4 ds_load_b64 B-fragments.
//
//    LDS layout is pair-interleaved so each B fragment {W[kk][d],W[kk+1][d]}
//    is one aligned ds_load_b64:  element (f,d) at ((f>>1)*64 + d)*2 + (f&1).
// ---------------------------------------------------------------------------
__launch_bounds__(128)
__global__ void gat_gemm_wmma(const float* __restrict__ x,
                              const float* __restrict__ W,   // [H, F, D]
                              float* __restrict__ Wh,        // [N, H*D]
                              int nnodes) {
    __shared__ float Blds[FIN * DHEAD];   // 64 KB, pair-interleaved W[h] slab

    const int h = blockIdx.y;
    const float4* Wg = (const float4*)(W + (size_t)h * FIN * DHEAD);
    for (int i = threadIdx.x; i < (FIN * DHEAD) / 4; i += blockDim.x) {
        const float4 wv = Wg[i];
        const int f = i >> 4;           // feature row 0..255
        const int d = (i & 15) << 2;    // col 0..60 step 4
        float* p = Blds + ((size_t)(f >> 1) * DHEAD + d) * 2 + (f & 1);
        p[0] = wv.x; p[2] = wv.y; p[4] = wv.z; p[6] = wv.w;
    }
    __syncthreads();

    const int wave  = threadIdx.x >> 5;
    const int lane  = threadIdx.x & 31;
    const int halfw = lane >> 4;        // 0: lanes 0-15, 1: lanes 16-31
    const int lm    = lane & 15;
    const int row0  = (blockIdx.x * 4 + wave) * 32;
    if (row0 >= nnodes) return;         // wave-uniform exit (EXEC stays all-1s)

    v8f acc0 = {}, acc1 = {}, acc2 = {}, acc3 = {};
    v8f acc4 = {}, acc5 = {}, acc6 = {}, acc7 = {};

    // A-fragment rows (two 16-row tiles); clamp for tail (stores are guarded)
    const int rowA0 = min(row0 + lm,      nnodes - 1);
    const int rowA1 = min(row0 + 16 + lm, nnodes - 1);
    const float* xrow0 = x + (size_t)rowA0 * FIN;
    const float* xrow1 = x + (size_t)rowA1 * FIN;
    __builtin_prefetch(xrow0, 0, 3);    // global_prefetch_b8
    __builtin_prefetch(xrow1, 0, 3);

    #pragma unroll 4
    for (int k = 0; k < FIN; k += 4) {
        // A 16x4 f32 frags: VGPR0 = A[m][kk], VGPR1 = A[m][kk+1], kk = k+2*half
        const int kk = k + 2 * halfw;   // always even
        const v2f a0 = *(const v2f*)(xrow0 + kk);
        const v2f a1 = *(const v2f*)(xrow1 + kk);

        // B 4x16 f32 frags: one aligned b64 each from pair-interleaved LDS
        const float* Bp = Blds + (size_t)(kk >> 1) * (DHEAD * 2);
        const v2f b0 = *(const v2f*)(Bp + 2 * lm);
        const v2f b1 = *(const v2f*)(Bp + 2 * (16 + lm));
        const v2f b2 = *(const v2f*)(Bp + 2 * (32 + lm));
        const v2f b3 = *(const v2f*)(Bp + 2 * (48 + lm));

        acc0 = __builtin_amdgcn_wmma_f32_16x16x4_f32(false, a0, false, b0, (short)0, acc0, false, false);
        acc1 = __builtin_amdgcn_wmma_f32_16x16x4_f32(false, a0, false, b1, (short)0, acc1, false, false);
        acc2 = __builtin_amdgcn_wmma_f32_16x16x4_f32(false, a0, false, b2, (short)0, acc2, false, false);
        acc3 = __builtin_amdgcn_wmma_f32_16x16x4_f32(false, a0, false, b3, (short)0, acc3, false, false);
        acc4 = __builtin_amdgcn_wmma_f32_16x16x4_f32(false, a1, false, b0, (short)0, acc4, false, false);
        acc5 = __builtin_amdgcn_wmma_f32_16x16x4_f32(false, a1, false, b1, (short)0, acc5, false, false);
        acc6 = __builtin_amdgcn_wmma_f32_16x16x4_f32(false, a1, false, b2, (short)0, acc6, false, false);
        acc7 = __builtin_amdgcn_wmma_f32_16x16x4_f32(false, a1, false, b3, (short)0, acc7, false, false);
    }

    // D 16x16 f32 layout: VGPR r -> row r (lanes 0-15) / row r+8 (lanes 16-31)
    const int colBase = h * DHEAD;
    #pragma unroll
    for (int r = 0; r < 8; ++r) {
        const int rowa = row0 + r + 8 * halfw;
        if (rowa < nnodes) {
            float* o = Wh + (size_t)rowa * COLS + colBase + lm;
            o[0]  = acc0[r];
            o[16] = acc1[r];
            o[32] = acc2[r];
            o[48] = acc3[r];
        }
        const int rowb = row0 + 16 + r + 8 * halfw;
        if (rowb < nnodes) {
            float* o = Wh + (size_t)rowb * COLS + colBase + lm;
            o[0]  = acc4[r];
            o[16] = acc5[r];
            o[32] = acc6[r];
            o[48] = acc7[r];
        }
    }
}

// ---------------------------------------------------------------------------
// 2) f1/f2 = Wh . a1/a2 + bias   (one thread per (n,h))
// ---------------------------------------------------------------------------
__global__ void gat_scores(const float* __restrict__ Wh,
                           const float* __restrict__ a1, const float* __restrict__ a1b,
                           const float* __restrict__ a2, const float* __restrict__ a2b,
                           float* __restrict__ f1, float* __restrict__ f2, int nnodes) {
    int idx = blockIdx.x * blockDim.x + threadIdx.x;
    if (idx >= nnodes * NHEADS) return;
    const int n = idx >> 2;
    const int h = idx & 3;
    const float4* w  = (const float4*)(Wh + (size_t)n * COLS + h * DHEAD);
    const float4* v1 = (const float4*)(a1 + h * DHEAD);
    const float4* v2 = (const float4*)(a2 + h * DHEAD);
    float s1 = 0.f, s2 = 0.f;
    #pragma unroll
    for (int i = 0; i < DHEAD / 4; ++i) {
        const float4 wv = w[i];
        const float4 x1 = v1[i];
        const float4 x2 = v2[i];
        s1 += wv.x * x1.x + wv.y * x1.y + wv.z * x1.z + wv.w * x1.w;
        s2 += wv.x * x2.x + wv.y * x2.y + wv.z * x2.z + wv.w * x2.w;
    }
    f1[idx] = s1 + a1b[h];
    f2[idx] = s2 + a2b[h];
}

// float atomic-max via signed-max / unsigned-min bit trick (init must be -inf)
__device__ __forceinline__ void atomicMaxFloat(float* addr, float val) {
    if (!(val < 0.0f))
        atomicMax((int*)addr, __float_as_int(val));
    else
        atomicMin((unsigned int*)addr, (unsigned int)__float_as_int(val));
}

// ---------------------------------------------------------------------------
// 3) edge logits -> leaky_relu -> segment max over src
// ---------------------------------------------------------------------------
__global__ void gat_edge_max(const int* __restrict__ src, const int* __restrict__ dst,
                             const float* __restrict__ bias,
                             const float* __restrict__ f1, const float* __restrict__ f2,
                             float* __restrict__ z, float* __restrict__ m, int nedges) {
    int e = blockIdx.x * blockDim.x + threadIdx.x;
    if (e >= nedges) return;
    const int i = src[e];
    const int j = dst[e];
    const float bv = bias[e];
    #pragma unroll
    for (int h = 0; h < NHEADS; ++h) {
        const float l  = bv * (f1[i * NHEADS + h] + f2[j * NHEADS + h]);
        const float zz = (l > 0.0f) ? l : 0.2f * l;   // leaky_relu(0.2)
        z[(size_t)e * NHEADS + h] = zz;
        atomicMaxFloat(&m[i * NHEADS + h], zz);
    }
}

// ---------------------------------------------------------------------------
// 4) e = exp(z - m[src]); denom = segment_sum  (z overwritten in place)
// ---------------------------------------------------------------------------
__global__ void gat_edge_exp(const int* __restrict__ src,
                             float* __restrict__ z, const float* __restrict__ m,
                             float* __restrict__ denom, int total) {
    int idx = blockIdx.x * blockDim.x + threadIdx.x;
    if (idx >= total) return;
    const int e = idx >> 2;
    const int h = idx & 3;
    const int i = src[e];
    float mv = m[i * NHEADS + h];
    mv = __builtin_isfinite(mv) ? mv : 0.0f;          // match reference
    const float ev = expf(z[idx] - mv);
    z[idx] = ev;
    atomicAdd(&denom[i * NHEADS + h], ev);
}

// ---------------------------------------------------------------------------
// 5) out[src] += alpha * Wh[dst]   (64 threads per edge, float4 each)
// ---------------------------------------------------------------------------
__global__ void gat_aggregate(const int* __restrict__ src, const int* __restrict__ dst,
                              const float* __restrict__ ez, const float* __restrict__ denom,
                              const float* __restrict__ Wh, float* __restrict__ out,
                              long total) {
    long idx = (long)blockIdx.x * blockDim.x + threadIdx.x;
    if (idx >= total) return;
    const int e  = (int)(idx >> 6);
    const int q  = (int)(idx & 63);
    const int h  = q >> 4;
    const int d4 = (q & 15) << 2;
    const int i = src[e];
    const int j = dst[e];
    const float alpha = ez[(size_t)e * NHEADS + h] /
                        fmaxf(denom[i * NHEADS + h], 1e-9f);
    const float4 wv = *(const float4*)(Wh + (size_t)j * COLS + h * DHEAD + d4);
    float* o = out + (size_t)i * COLS + h * DHEAD + d4;
    atomicAdd(o + 0, alpha * wv.x);
    atomicAdd(o + 1, alpha * wv.y);
    atomicAdd(o + 2, alpha * wv.z);
    atomicAdd(o + 3, alpha * wv.w);
}

// ---------------------------------------------------------------------------
// 6) out = elu(out + out_b)
// ---------------------------------------------------------------------------
__global__ void gat_finalize(float* __restrict__ out, const float* __restrict__ outb,
                             long total) {
    long idx = (long)blockIdx.x * blockDim.x + threadIdx.x;
    if (idx >= total) return;
    const float v = out[idx] + outb[idx & (COLS - 1)];
    out[idx] = (v > 0.0f) ? v : (expf(v) - 1.0f);
}

// ---------------------------------------------------------------------------
extern "C" void kernel_launch(void* const* d_in, const int* in_sizes, int n_in,
                              void* d_out, int out_size, void* d_ws, size_t ws_size,
                              hipStream_t stream) {
    const float* x    = (const float*)d_in[0];
    const int*   ei   = (const int*)d_in[1];
    const float* bias = (const float*)d_in[2];
    const float* W    = (const float*)d_in[3];
    const float* a1   = (const float*)d_in[4];
    const float* a1b  = (const float*)d_in[5];
    const float* a2   = (const float*)d_in[6];
    const float* a2b  = (const float*)d_in[7];
    const float* outb = (const float*)d_in[8];
    float* out = (float*)d_out;

    const int n = in_sizes[0] / FIN;   // N nodes
    const int e = in_sizes[2];         // E edges
    const int* src = ei;               // edge_index[0]
    const int* dst = ei + e;           // edge_index[1]

    // workspace layout (floats)
    float* ws = (float*)d_ws;
    size_t off = 0;
    float* Wh = ws + off; off += (size_t)n * COLS;
    float* f1 = ws + off; off += (size_t)n * NHEADS;
    float* f2 = ws + off; off += (size_t)n * NHEADS;
    float* m  = ws + off; off += (size_t)n * NHEADS;
    float* dn = ws + off; off += (size_t)n * NHEADS;
    float* z  = ws + off; off += (size_t)e * NHEADS;
    (void)ws_size;

    {   // init accumulators / reductions
        const long total = (long)n * COLS;
        gat_init<<<(int)((total + 255) / 256), 256, 0, stream>>>(out, m, dn, total,
                                                                 n * NHEADS);
    }
    {   // fp32 WMMA projection (32 rows per wave)
        dim3 grid((n + 127) / 128, NHEADS);
        gat_gemm_wmma<<<grid, 128, 0, stream>>>(x, W, Wh, n);
    }
    {   // attention scores
        const int total = n * NHEADS;
        gat_scores<<<(total + 255) / 256, 256, 0, stream>>>(Wh, a1, a1b, a2, a2b,
                                                            f1, f2, n);
    }
    gat_edge_max<<<(e + 255) / 256, 256, 0, stream>>>(src, dst, bias, f1, f2, z, m, e);
    {   // exp + denom
        const int total = e * NHEADS;
        gat_edge_exp<<<(total + 255) / 256, 256, 0, stream>>>(src, z, m, dn, total);
    }
    {   // message aggregation
        const long total = (long)e * 64;
        gat_aggregate<<<(int)((total + 255) / 256), 256, 0, stream>>>(src, dst, z, dn,
                                                                      Wh, out, total);
    }
    {   // bias + ELU
        const long total = (long)n * COLS;
        gat_finalize<<<(int)((total + 255) / 256), 256, 0, stream>>>(out, outb, total);
    }
}